// GNN_Model_58884001628362
// MI455X (gfx1250) — compile-verified
//
#include <hip/hip_runtime.h>
#include <math.h>

#define TB  256      // threads per block
#define EPB 1024     // edges staged in LDS per block

#if __has_builtin(__builtin_amdgcn_global_load_async_to_lds_b32)
#define USE_ASYNC_LDS 1
#else
#define USE_ASYNC_LDS 0
#endif

// ---------------------------------------------------------------------------
// HOT KERNEL (placed first so the disasm snippet shows it).
// Edge scatter: acc[dst] += dis[src]*dis[dst] * z[src]
// Edge indices staged into LDS via gfx1250 async copy; drained with
// s_wait_asynccnt before the block barrier.
// ---------------------------------------------------------------------------
__global__ void __launch_bounds__(TB)
scatter(const int* __restrict__ src, const int* __restrict__ dst,
        const float* __restrict__ dis, const float2* __restrict__ z,
        float* __restrict__ acc, int E) {
    __shared__ int s_src[EPB];
    __shared__ int s_dst[EPB];
    const int tid = threadIdx.x;
    for (int base = blockIdx.x * EPB; base < E; base += gridDim.x * EPB) {
        int n = E - base; if (n > EPB) n = EPB;
#if USE_ASYNC_LDS
        for (int k = tid; k < n; k += TB) {
            __builtin_amdgcn_global_load_async_to_lds_b32(
                (int*)(src + base + k), (int*)&s_src[k], 0, 0);
            __builtin_amdgcn_global_load_async_to_lds_b32(
                (int*)(dst + base + k), (int*)&s_dst[k], 0, 0);
        }
#if __has_builtin(__builtin_amdgcn_s_wait_asynccnt)
        __builtin_amdgcn_s_wait_asynccnt(0);
#else
        asm volatile("s_wait_asynccnt 0" ::: "memory");
#endif
#else
        for (int k = tid; k < n; k += TB) {
            if (base + k + 2 * EPB < E) __builtin_prefetch(src + base + k + 2 * EPB, 0, 1);
            s_src[k] = src[base + k];
            s_dst[k] = dst[base + k];
        }
#endif
        __syncthreads();
        for (int k = tid; k < n; k += TB) {
            int s = s_src[k];
            int t = s_dst[k];
            float w = dis[s] * dis[t];
            float2 zz = z[s];
            unsafeAtomicAdd(&acc[2 * t],     w * zz.x);   // global_atomic_add_f32 (no return)
            unsafeAtomicAdd(&acc[2 * t + 1], w * zz.y);
        }
        __syncthreads();
    }
}

// ---------------------------------------------------------------------------
// Per-node transform: z = x@Wb + cb ; acc = x@Wa + ca + dis^2 * z  (self term)
// ---------------------------------------------------------------------------
__global__ void __launch_bounds__(TB)
transform(const float2* __restrict__ xin, int relu_in,
          const float* __restrict__ cons,
          const float* __restrict__ dis,
          float2* __restrict__ z, float2* __restrict__ acc, int N) {
    int i = blockIdx.x * TB + threadIdx.x;
    if (i >= N) return;
    const float wa00 = cons[0], wa01 = cons[1], wa10 = cons[2], wa11 = cons[3];
    const float wb00 = cons[4], wb01 = cons[5], wb10 = cons[6], wb11 = cons[7];
    const float ca0 = cons[8], ca1 = cons[9], cb0 = cons[10], cb1 = cons[11];
    float2 x = xin[i];
    float x0 = x.x, x1 = x.y;
    if (relu_in) { x0 = fmaxf(x0, 0.0f); x1 = fmaxf(x1, 0.0f); }
    float z0 = fmaf(x1, wb10, fmaf(x0, wb00, cb0));
    float z1 = fmaf(x1, wb11, fmaf(x0, wb01, cb1));
    float d = dis[i];
    float sw = d * d;
    float a0 = fmaf(sw, z0, fmaf(x1, wa10, fmaf(x0, wa00, ca0)));
    float a1 = fmaf(sw, z1, fmaf(x1, wa11, fmaf(x0, wa01, ca1)));
    z[i]   = make_float2(z0, z1);
    acc[i] = make_float2(a0, a1);
}

// ---------------------------------------------------------------------------
// Fold all small weights into per-layer constants:
//   per layer 12 floats: Wa[4] (r*2+c), Wb[4], ca[2], cb[2]
//   h = x@Wa + ca + prop(x@Wb + cb),   x_next = relu(h)
// Layer 0 absorbs the step-embedding concat and both biases.
// ---------------------------------------------------------------------------
__global__ void prep_consts(const float* __restrict__ W0, const float* __restrict__ b0,
                            const float* __restrict__ Wh, const float* __restrict__ bh,
                            const float* __restrict__ step_emb,
                            const int* __restrict__ step_index,
                            float* __restrict__ cons) {
    if (threadIdx.x != 0 || blockIdx.x != 0) return;
    const float* s = step_emb + step_index[0] * 8;
    // layer 0: W0 shape (2,10,2)
    for (int r = 0; r < 2; ++r)
        for (int c = 0; c < 2; ++c) {
            cons[r*2 + c]     = W0[r*2 + c];        // Wa = W0[0][0:2]
            cons[4 + r*2 + c] = W0[20 + r*2 + c];   // Wb = W0[1][0:2]
        }
    for (int c = 0; c < 2; ++c) {
        float ca = b0[c] + b0[2 + c];
        float cb = 0.0f;
        for (int j = 0; j < 8; ++j) {
            ca += s[j] * W0[(2 + j)*2 + c];         // s @ W0[0][2:]
            cb += s[j] * W0[20 + (2 + j)*2 + c];    // s @ W0[1][2:]
        }
        cons[8 + c]  = ca;
        cons[10 + c] = cb;
    }
    // layers 1..7: Wh shape (7,2,2,2), bh shape (7,2,2)
    for (int l = 1; l < 8; ++l) {
        const float* W = Wh + (l - 1) * 8;
        const float* b = bh + (l - 1) * 4;
        float* cl = cons + l * 12;
        for (int i = 0; i < 4; ++i) { cl[i] = W[i]; cl[4 + i] = W[4 + i]; }
        cl[8]  = b[0] + b[2];
        cl[9]  = b[1] + b[3];
        cl[10] = 0.0f;
        cl[11] = 0.0f;
    }
}

__global__ void __launch_bounds__(TB)
deg_init(unsigned* __restrict__ deg, int N) {
    int i = blockIdx.x * TB + threadIdx.x;
    if (i < N) deg[i] = 1u;                          // self-loop
}

__global__ void __launch_bounds__(TB)
deg_count(const int* __restrict__ dst, unsigned* __restrict__ deg, int E) {
    int e = blockIdx.x * TB + threadIdx.x;
    if (e < E) {
        if (e + 8 * TB < E) __builtin_prefetch(dst + e + 8 * TB, 0, 1); // global_prefetch_b8
        atomicAdd(&deg[dst[e]], 1u);
    }
}

__global__ void __launch_bounds__(TB)
dis_kernel(const unsigned* __restrict__ deg, float* __restrict__ dis, int N) {
    int i = blockIdx.x * TB + threadIdx.x;
    if (i < N) dis[i] = 1.0f / sqrtf((float)deg[i]);
}

__global__ void __launch_bounds__(TB)
finalize(const float2* __restrict__ acc, float2* __restrict__ out, int N) {
    int i = blockIdx.x * TB + threadIdx.x;
    if (i < N) {
        float2 a = acc[i];
        out[i] = make_float2(fmaxf(a.x, 0.0f), fmaxf(a.y, 0.0f));
    }
}

// ---------------------------------------------------------------------------
extern "C" void kernel_launch(void* const* d_in, const int* in_sizes, int n_in,
                              void* d_out, int out_size, void* d_ws, size_t ws_size,
                              hipStream_t stream) {
    const float* X          = (const float*)d_in[0];
    const int*   edge_index = (const int*)  d_in[1];
    const int*   step_index = (const int*)  d_in[2];
    const float* step_emb   = (const float*)d_in[3];
    const float* W0         = (const float*)d_in[4];
    const float* b0         = (const float*)d_in[5];
    const float* Wh         = (const float*)d_in[6];
    const float* bh         = (const float*)d_in[7];
    (void)n_in; (void)out_size; (void)ws_size;

    const int N = in_sizes[0] / 2;
    const int E = in_sizes[1] / 2;
    const int* src = edge_index;        // row 0
    const int* dst = edge_index + E;    // row 1

    char* ws = (char*)d_ws;
    size_t off = 0;
    auto ws_alloc = [&](size_t bytes) -> void* {
        void* p = (void*)(ws + off);
        off += (bytes + 255) & ~(size_t)255;
        return p;
    };
    unsigned* deg  = (unsigned*)ws_alloc((size_t)N * 4);
    float*    dis  = (float*)   ws_alloc((size_t)N * 4);
    float2*   z    = (float2*)  ws_alloc((size_t)N * 8);
    float2*   accA = (float2*)  ws_alloc((size_t)N * 8);
    float2*   accB = (float2*)  ws_alloc((size_t)N * 8);
    float*    cons = (float*)   ws_alloc(96 * 4);

    const int nblkN = (N + TB - 1) / TB;
    const int nblkE = (E + TB - 1) / TB;
    const int nblkS = (E + EPB - 1) / EPB;

    prep_consts<<<1, 1, 0, stream>>>(W0, b0, Wh, bh, step_emb, step_index, cons);
    deg_init  <<<nblkN, TB, 0, stream>>>(deg, N);
    deg_count <<<nblkE, TB, 0, stream>>>(dst, deg, E);
    dis_kernel<<<nblkN, TB, 0, stream>>>(deg, dis, N);

    const float2* xin = (const float2*)X;
    float2* bufs[2] = {accA, accB};
    for (int l = 0; l < 8; ++l) {
        float2* acc = bufs[l & 1];
        transform<<<nblkN, TB, 0, stream>>>(xin, (l > 0) ? 1 : 0, cons + l * 12, dis, z, acc, N);
        scatter  <<<nblkS, TB, 0, stream>>>(src, dst, dis, z, (float*)acc, E);
        xin = acc;
    }
    finalize<<<nblkN, TB, 0, stream>>>(xin, (float2*)d_out, N);
}